// SGCLayer_10531259810063
// MI455X (gfx1250) — compile-verified
//
#include <hip/hip_runtime.h>

#define IN_F  256
#define OUT_F 128
#define LDS_PAD 260   // dword row stride: even (8B-aligned b64 LDS loads), %64==4 (no bank conflicts)

typedef __attribute__((ext_vector_type(2))) float v2f;
typedef __attribute__((ext_vector_type(8))) float v8f;

// ---------------------------------------------------------------------------
// GEMM: x[N,128] = h[N,256] @ w[256,128], fp32 via V_WMMA_F32_16X16X4_F32.
// Block = 256 threads (8 wave32) handles a 16-row strip; wave w owns cols
// [16w,16w+16). A tile staged in padded LDS; 64 WMMAs per wave over K=256.
// ---------------------------------------------------------------------------
__global__ __launch_bounds__(256) void sgc_gemm_wmma(const float* __restrict__ h,
                                                     const float* __restrict__ w,
                                                     float* __restrict__ x) {
  __shared__ float atile[16 * LDS_PAD];
  const int m0  = blockIdx.x * 16;
  const int tid = threadIdx.x;

  // Cooperative load of the 16x256 fp32 A tile (float4, fully coalesced).
  for (int i4 = tid; i4 < 16 * IN_F / 4; i4 += 256) {
    int i = i4 * 4;
    int r = i / IN_F, c = i % IN_F;
    float4 v = *(const float4*)(h + (size_t)(m0 + r) * IN_F + c);
    atile[r * LDS_PAD + c + 0] = v.x;
    atile[r * LDS_PAD + c + 1] = v.y;
    atile[r * LDS_PAD + c + 2] = v.z;
    atile[r * LDS_PAD + c + 3] = v.w;
  }
  __syncthreads();

  const int wave = tid >> 5;
  const int lane = tid & 31;
  const int half = lane >> 4;   // lanes 16-31 hold K+2,K+3 per 32-bit A/B layout
  const int l    = lane & 15;
  const int n0   = wave * 16;

  v8f c = {};
  const float* arow = atile + l * LDS_PAD + 2 * half;          // A: M=l, K=k+2*half..+1
  const float* bcol = w + (size_t)(2 * half) * OUT_F + n0 + l; // B: N=n0+l, K=k+2*half..+1
  for (int k = 0; k < IN_F; k += 4) {
    float2 av = *(const float2*)(arow + k);                    // ds_load_b64, conflict-free
    v2f a, b;
    a.x = av.x;
    a.y = av.y;
    b.x = bcol[(size_t)k * OUT_F];
    b.y = bcol[(size_t)(k + 1) * OUT_F];
    c = __builtin_amdgcn_wmma_f32_16x16x4_f32(false, a, false, b, (short)0, c,
                                              false, false);
  }

  // C/D layout: VGPR r -> (M = r + 8*half, N = l)
  float* out = x + (size_t)(m0 + 8 * half) * OUT_F + n0 + l;
#pragma unroll
  for (int r = 0; r < 8; ++r) out[(size_t)r * OUT_F] = c[r];
}

// ---------------------------------------------------------------------------
// Edge scatter: one wave per edge. Lane j gathers float4 of x[src] (b128,
// coalesced, L2-resident: 51.2MB < 192MB L2), scales by norm[src], and does 4
// non-returning global_atomic_add_f32 into acc[dst] (L2 atomic units).
// ---------------------------------------------------------------------------
__global__ __launch_bounds__(256) void sgc_scatter(const float* __restrict__ x,
                                                   const float* __restrict__ norm,
                                                   const int* __restrict__ src,
                                                   const int* __restrict__ dst,
                                                   float* __restrict__ acc, int E) {
  int gw   = (blockIdx.x * 256 + threadIdx.x) >> 5;
  int lane = threadIdx.x & 31;
  if (gw >= E) return;
  int s = __builtin_amdgcn_readfirstlane(src[gw]);  // wave-uniform -> SGPR
  int d = __builtin_amdgcn_readfirstlane(dst[gw]);
  float ns = norm[s];
  float4 v = *(const float4*)(x + (size_t)s * OUT_F + lane * 4);
  float r0 = ns * v.x, r1 = ns * v.y, r2 = ns * v.z, r3 = ns * v.w;
  float* a = acc + (size_t)d * OUT_F + lane * 4;
  // Non-returning f32 adds (TH=0): tracked on STOREcnt, implicit wait at endpgm.
  asm volatile(
      "global_atomic_add_f32 %4, %0, off\n\t"
      "global_atomic_add_f32 %4, %1, off offset:4\n\t"
      "global_atomic_add_f32 %4, %2, off offset:8\n\t"
      "global_atomic_add_f32 %4, %3, off offset:12"
      :
      : "v"(r0), "v"(r1), "v"(r2), "v"(r3), "v"(a)
      : "memory");
}

// acc := 0 (accumulator init before first hop)
__global__ __launch_bounds__(256) void sgc_zero(float4* __restrict__ p, int total4) {
  int i = blockIdx.x * 256 + threadIdx.x;
  if (i < total4) p[i] = make_float4(0.f, 0.f, 0.f, 0.f);
}

// xout := norm[node] * acc ; acc := 0   (fuses dst-norm scale with re-zeroing,
// saving a full 51MB memset pass between hops)
__global__ __launch_bounds__(256) void sgc_scale_zero(const float* __restrict__ norm,
                                                      float* __restrict__ acc,
                                                      float* __restrict__ xout,
                                                      int total4) {
  int i4 = blockIdx.x * 256 + threadIdx.x;
  if (i4 >= total4) return;
  int node = i4 >> 5;  // (i4*4)/128
  float nv = norm[node];
  float4 v = ((const float4*)acc)[i4];
  ((float4*)xout)[i4] = make_float4(nv * v.x, nv * v.y, nv * v.z, nv * v.w);
  ((float4*)acc)[i4] = make_float4(0.f, 0.f, 0.f, 0.f);
}

extern "C" void kernel_launch(void* const* d_in, const int* in_sizes, int n_in,
                              void* d_out, int out_size, void* d_ws, size_t ws_size,
                              hipStream_t stream) {
  const float* h    = (const float*)d_in[0];
  const float* w    = (const float*)d_in[1];
  const float* norm = (const float*)d_in[2];
  const int*   src  = (const int*)d_in[3];
  const int*   dst  = (const int*)d_in[4];
  float* out = (float*)d_out;
  float* acc = (float*)d_ws;  // N*128 fp32 accumulator (51.2 MB)

  const int N = in_sizes[2];  // norm has N elements
  const int E = in_sizes[3];

  const int mblocks = (N + 15) / 16;               // N=100000 -> 6250 (exact)
  const int total4  = N * OUT_F / 4;
  const int sblocks = (total4 + 255) / 256;
  const int eblocks = (E + 7) / 8;                 // 8 edges (waves) per block

  // x0 = h @ W  (into d_out; reused as the per-hop feature buffer)
  sgc_gemm_wmma<<<mblocks, 256, 0, stream>>>(h, w, out);
  sgc_zero<<<sblocks, 256, 0, stream>>>((float4*)acc, total4);

  // hop 1: acc += norm[src]*x[src] per edge; x = norm*acc; acc = 0
  sgc_scatter<<<eblocks, 256, 0, stream>>>(out, norm, src, dst, acc, E);
  sgc_scale_zero<<<sblocks, 256, 0, stream>>>(norm, acc, out, total4);

  // hop 2
  sgc_scatter<<<eblocks, 256, 0, stream>>>(out, norm, src, dst, acc, E);
  sgc_scale_zero<<<sblocks, 256, 0, stream>>>(norm, acc, out, total4);
}